// CommunityPreservationLoss_10754598109319
// MI455X (gfx1250) — compile-verified
//
#include <hip/hip_runtime.h>
#include <hip/hip_bf16.h>

#define N_NODES    8192
#define EMBED_DIM  128
#define MARGIN_F   1.0f
#define TILE       16
#define KSTEPS     (EMBED_DIM / 4)        // 32 wmma_f32_16x16x4 steps per tile
#define NTILES     (N_NODES / TILE)       // 512
#define ITPW       2                      // i-tiles per wave (rows = 32)
#define IPAIRS     (NTILES / ITPW)        // 256
#define JTPW       64                     // j-tiles per wave
#define JCHUNKS    (NTILES / JTPW)        // 8
#define INF_BITS   0x7F800000u
#define INFF       __builtin_inff()

typedef float v2f __attribute__((ext_vector_type(2)));
typedef float v8f __attribute__((ext_vector_type(8)));

// ---------------------------------------------------------------------------
// Helpers: batched B-tile load, dual-chain WMMA GEMM (2 i-tiles vs 1 j-tile)
// ---------------------------------------------------------------------------
__device__ __forceinline__ void load_btile(const float* __restrict__ bbase,
                                           v2f (&b)[KSTEPS]) {
  #pragma unroll
  for (int s = 0; s < KSTEPS; s++) b[s] = *(const v2f*)(bbase + 4 * s);
}

__device__ __forceinline__ void gemm2(const v2f (&a0)[KSTEPS],
                                      const v2f (&a1)[KSTEPS],
                                      const v2f (&b)[KSTEPS],
                                      v8f& c0, v8f& c1) {
  v8f x0 = {}, x1 = {};
  #pragma unroll
  for (int s = 0; s < KSTEPS; s++) {
    x0 = __builtin_amdgcn_wmma_f32_16x16x4_f32(false, a0[s], false, b[s],
                                               (short)0, x0, false, false);
    x1 = __builtin_amdgcn_wmma_f32_16x16x4_f32(false, a1[s], false, b[s],
                                               (short)0, x1, false, false);
  }
  c0 = x0; c1 = x1;
}

// ---------------------------------------------------------------------------
// Kernel 1: per-row sum of squares (one wave per row) + workspace init.
// ---------------------------------------------------------------------------
__global__ __launch_bounds__(256) void k_rowsq_init(
    const float* __restrict__ x, float* __restrict__ sq,
    unsigned* __restrict__ minneg, float* __restrict__ sum,
    unsigned* __restrict__ cnt) {
  int wave = (int)((blockIdx.x * blockDim.x + threadIdx.x) >> 5);
  int lane = (int)(threadIdx.x & 31);
  if (wave < N_NODES) {
    const float4* row = (const float4*)(x + (size_t)wave * EMBED_DIM);
    float4 v = row[lane];                          // 32 lanes * 4 = 128 elems
    float s = v.x * v.x + v.y * v.y + v.z * v.z + v.w * v.w;
    #pragma unroll
    for (int m = 16; m >= 1; m >>= 1) s += __shfl_xor(s, m, 32);
    if (lane == 0) sq[wave] = s;
    if (lane == 1) minneg[wave] = INF_BITS;
  }
  if (blockIdx.x == 0 && threadIdx.x == 0) { *sum = 0.0f; *cnt = 0u; }
}

// ---------------------------------------------------------------------------
// Pass A: hardest-negative (min d^2 over other-community columns) per row.
// One wave owns 2 anchor tiles (32 rows) and streams 64 column tiles,
// double-buffering the B fragments.
// ---------------------------------------------------------------------------
__device__ __forceinline__ void epiA(int jb, int l16,
                                     const int (&ci)[16], const float (&sqi)[16],
                                     const v8f& c0, const v8f& c1,
                                     const int* __restrict__ comm,
                                     const float* __restrict__ sq,
                                     float (&rowmin)[16]) {
  int   jg  = jb + l16;
  int   cj  = comm[jg];
  float sqj = sq[jg];
  #pragma unroll
  for (int r = 0; r < 8; r++) {
    float d2 = fmaxf(sqi[r] + sqj - 2.0f * c0[r], 0.0f);
    rowmin[r] = fminf(rowmin[r], (cj != ci[r]) ? d2 : INFF);
    float e2 = fmaxf(sqi[8 + r] + sqj - 2.0f * c1[r], 0.0f);
    rowmin[8 + r] = fminf(rowmin[8 + r], (cj != ci[8 + r]) ? e2 : INFF);
  }
}

__global__ __launch_bounds__(256) void k_minneg(
    const float* __restrict__ x, const int* __restrict__ comm,
    const float* __restrict__ sq, unsigned* __restrict__ minneg) {
  int wave  = (int)((blockIdx.x * blockDim.x + threadIdx.x) >> 5);
  int lane  = (int)(threadIdx.x & 31);
  int pair  = wave / JCHUNKS;
  int chunk = wave % JCHUNKS;
  int i0    = pair * (ITPW * TILE);          // 32 anchor rows
  int j0    = chunk * (JTPW * TILE);
  int half  = lane >> 4;
  int l16   = lane & 15;
  int koff  = 2 * half;

  // Resident A fragments for both anchor tiles (2 x 64 VGPRs).
  v2f a0[KSTEPS], a1[KSTEPS];
  load_btile(x + (size_t)(i0 + l16) * EMBED_DIM + koff, a0);
  load_btile(x + (size_t)(i0 + TILE + l16) * EMBED_DIM + koff, a1);

  // Row metadata: index t*8+r -> row i0 + 16*t + 8*half + r.
  int   ci[16];
  float sqi[16];
  float rowmin[16];
  #pragma unroll
  for (int t = 0; t < 2; t++)
    #pragma unroll
    for (int r = 0; r < 8; r++) {
      int row = i0 + 16 * t + 8 * half + r;
      ci[t * 8 + r]     = comm[row];
      sqi[t * 8 + r]    = sq[row];
      rowmin[t * 8 + r] = INFF;
    }

  v2f b0[KSTEPS], b1[KSTEPS];
  load_btile(x + (size_t)(j0 + l16) * EMBED_DIM + koff, b0);

  for (int jt = 0; jt < JTPW - 2; jt += 2) {
    load_btile(x + (size_t)(j0 + (jt + 1) * TILE + l16) * EMBED_DIM + koff, b1);
    { v8f c0, c1; gemm2(a0, a1, b0, c0, c1);
      epiA(j0 + jt * TILE, l16, ci, sqi, c0, c1, comm, sq, rowmin); }
    load_btile(x + (size_t)(j0 + (jt + 2) * TILE + l16) * EMBED_DIM + koff, b0);
    { v8f c0, c1; gemm2(a0, a1, b1, c0, c1);
      epiA(j0 + (jt + 1) * TILE, l16, ci, sqi, c0, c1, comm, sq, rowmin); }
  }
  { // peeled tail: tiles JTPW-2, JTPW-1
    load_btile(x + (size_t)(j0 + (JTPW - 1) * TILE + l16) * EMBED_DIM + koff, b1);
    { v8f c0, c1; gemm2(a0, a1, b0, c0, c1);
      epiA(j0 + (JTPW - 2) * TILE, l16, ci, sqi, c0, c1, comm, sq, rowmin); }
    { v8f c0, c1; gemm2(a0, a1, b1, c0, c1);
      epiA(j0 + (JTPW - 1) * TILE, l16, ci, sqi, c0, c1, comm, sq, rowmin); }
  }

  // Min across the 16 lanes of each half, then one atomicMin per row
  // (uint encoding is order-preserving for d^2 >= 0).
  #pragma unroll
  for (int t = 0; t < 2; t++)
    #pragma unroll
    for (int r = 0; r < 8; r++) {
      float v = rowmin[t * 8 + r];
      #pragma unroll
      for (int m = 8; m >= 1; m >>= 1) v = fminf(v, __shfl_xor(v, m, 32));
      if (l16 == 0)
        atomicMin(&minneg[i0 + 16 * t + 8 * half + r], __float_as_uint(v));
    }
}

// ---------------------------------------------------------------------------
// Pass B: hinge over positive pairs; sum + count accumulated per wave.
// ---------------------------------------------------------------------------
__device__ __forceinline__ void epiB(int jb, int i0, int half, int l16,
                                     const int (&ci)[16], const float (&sqi)[16],
                                     const float (&mneg)[16], const bool (&hneg)[16],
                                     const v8f& c0, const v8f& c1,
                                     const int* __restrict__ comm,
                                     const float* __restrict__ sq,
                                     float& lsum, unsigned& lcnt) {
  int   jg  = jb + l16;
  int   cj  = comm[jg];
  float sqj = sq[jg];
  #pragma unroll
  for (int t = 0; t < 2; t++)
    #pragma unroll
    for (int r = 0; r < 8; r++) {
      int   idx = t * 8 + r;
      int   ig  = i0 + 16 * t + 8 * half + r;
      float g   = (t == 0) ? c0[r] : c1[r];
      if ((cj == ci[idx]) && (jg != ig) && hneg[idx]) {
        float d2 = fmaxf(sqi[idx] + sqj - 2.0f * g, 0.0f);
        float d  = __builtin_sqrtf(d2);
        lsum += fmaxf(d - mneg[idx] + MARGIN_F, 0.0f);
        lcnt += 1u;
      }
    }
}

__global__ __launch_bounds__(256) void k_possum(
    const float* __restrict__ x, const int* __restrict__ comm,
    const float* __restrict__ sq, const unsigned* __restrict__ minneg,
    float* __restrict__ sum, unsigned* __restrict__ cnt) {
  int wave  = (int)((blockIdx.x * blockDim.x + threadIdx.x) >> 5);
  int lane  = (int)(threadIdx.x & 31);
  int pair  = wave / JCHUNKS;
  int chunk = wave % JCHUNKS;
  int i0    = pair * (ITPW * TILE);
  int j0    = chunk * (JTPW * TILE);
  int half  = lane >> 4;
  int l16   = lane & 15;
  int koff  = 2 * half;

  v2f a0[KSTEPS], a1[KSTEPS];
  load_btile(x + (size_t)(i0 + l16) * EMBED_DIM + koff, a0);
  load_btile(x + (size_t)(i0 + TILE + l16) * EMBED_DIM + koff, a1);

  int   ci[16];
  float sqi[16];
  float mneg[16];
  bool  hneg[16];
  #pragma unroll
  for (int t = 0; t < 2; t++)
    #pragma unroll
    for (int r = 0; r < 8; r++) {
      int row = i0 + 16 * t + 8 * half + r;
      int idx = t * 8 + r;
      ci[idx]  = comm[row];
      sqi[idx] = sq[row];
      unsigned bits = minneg[row];
      hneg[idx] = (bits != INF_BITS);
      mneg[idx] = __builtin_sqrtf(__uint_as_float(bits));
    }

  float    lsum = 0.0f;
  unsigned lcnt = 0u;

  v2f b0[KSTEPS], b1[KSTEPS];
  load_btile(x + (size_t)(j0 + l16) * EMBED_DIM + koff, b0);

  for (int jt = 0; jt < JTPW - 2; jt += 2) {
    load_btile(x + (size_t)(j0 + (jt + 1) * TILE + l16) * EMBED_DIM + koff, b1);
    { v8f c0, c1; gemm2(a0, a1, b0, c0, c1);
      epiB(j0 + jt * TILE, i0, half, l16, ci, sqi, mneg, hneg, c0, c1, comm, sq, lsum, lcnt); }
    load_btile(x + (size_t)(j0 + (jt + 2) * TILE + l16) * EMBED_DIM + koff, b0);
    { v8f c0, c1; gemm2(a0, a1, b1, c0, c1);
      epiB(j0 + (jt + 1) * TILE, i0, half, l16, ci, sqi, mneg, hneg, c0, c1, comm, sq, lsum, lcnt); }
  }
  {
    load_btile(x + (size_t)(j0 + (JTPW - 1) * TILE + l16) * EMBED_DIM + koff, b1);
    { v8f c0, c1; gemm2(a0, a1, b0, c0, c1);
      epiB(j0 + (JTPW - 2) * TILE, i0, half, l16, ci, sqi, mneg, hneg, c0, c1, comm, sq, lsum, lcnt); }
    { v8f c0, c1; gemm2(a0, a1, b1, c0, c1);
      epiB(j0 + (JTPW - 1) * TILE, i0, half, l16, ci, sqi, mneg, hneg, c0, c1, comm, sq, lsum, lcnt); }
  }

  #pragma unroll
  for (int m = 16; m >= 1; m >>= 1) {
    lsum += __shfl_xor(lsum, m, 32);
    lcnt += __shfl_xor(lcnt, m, 32);
  }
  if (lane == 0) {
    atomicAdd(sum, lsum);
    atomicAdd(cnt, lcnt);
  }
}

// ---------------------------------------------------------------------------
// Finalize: mean over valid triplets (0 if none).
// ---------------------------------------------------------------------------
__global__ void k_final(const float* __restrict__ sum,
                        const unsigned* __restrict__ cnt,
                        float* __restrict__ out) {
  if (threadIdx.x == 0 && blockIdx.x == 0) {
    unsigned c = *cnt;
    out[0] = (c > 0u) ? (*sum / (float)c) : 0.0f;
  }
}

// ---------------------------------------------------------------------------
extern "C" void kernel_launch(void* const* d_in, const int* in_sizes, int n_in,
                              void* d_out, int out_size, void* d_ws, size_t ws_size,
                              hipStream_t stream) {
  const float* x    = (const float*)d_in[0];   // embeddings (8192, 128) f32
  const int*   comm = (const int*)d_in[1];     // communities (8192,) int
  float*       out  = (float*)d_out;           // scalar f32

  char*     ws     = (char*)d_ws;
  float*    sq     = (float*)ws;                                   // 32 KB
  unsigned* minneg = (unsigned*)(ws + (size_t)N_NODES * 4);        // 32 KB
  float*    sum    = (float*)(ws + (size_t)2 * N_NODES * 4);
  unsigned* cnt    = (unsigned*)(ws + (size_t)2 * N_NODES * 4 + 4);

  // 1 wave per row: 8192 waves, 8 waves/block
  k_rowsq_init<<<N_NODES / 8, 256, 0, stream>>>(x, sq, minneg, sum, cnt);

  // 256 i-tile-pairs x 8 j-chunks = 2048 waves, 8 waves/block -> 256 blocks
  const int gemm_blocks = (IPAIRS * JCHUNKS) / 8;
  k_minneg<<<gemm_blocks, 256, 0, stream>>>(x, comm, sq, minneg);
  k_possum<<<gemm_blocks, 256, 0, stream>>>(x, comm, sq, minneg, sum, cnt);

  k_final<<<1, 32, 0, stream>>>(sum, cnt, out);
}